// ClassificationHead_67568425500679
// MI455X (gfx1250) — compile-verified
//
#include <hip/hip_runtime.h>

// ---------------------------------------------------------------------------
// ClassificationHead on MI455X (gfx1250): fused trim+mask -> fp32 WMMA GEMM
//   x:    (B=8, S=4096, H=1024) fp32
//   mask: (B, S) int32, contiguous prefix of ones
//   W:    (L=128, H=1024) fp32
//   b:    (L,) fp32
// Outputs (concatenated in d_out):
//   logits:  (B, S-2, L)  fp32   logits = context @ W^T + b
//   context: (B, S-2, H)  fp32   masked/trimmed x
// GEMM: M=32752, N=128, K=1024 via V_WMMA_F32_16X16X4_F32 (fp32 end-to-end).
// ---------------------------------------------------------------------------

typedef __attribute__((ext_vector_type(2))) float v2f;
typedef __attribute__((ext_vector_type(8))) float v8f;

#define BSZ   8
#define SDIM  4096
#define HDIM  1024
#define LDIM  128
#define SM2   (SDIM - 2)            // 4094 trimmed rows per batch
#define MROWS (BSZ * SM2)           // 32752 total GEMM rows (multiple of 16)
#define MTILE 64                    // rows per workgroup (4 x 16-row tiles)

// --- per-batch sequence length from the prefix-of-ones mask -----------------
__global__ __launch_bounds__(256)
void seqlen_kernel(const int* __restrict__ mask, int* __restrict__ len) {
    __shared__ int red[256];
    const int b = blockIdx.x;
    const int t = threadIdx.x;
    int s = 0;
    for (int i = t; i < SDIM; i += 256) s += mask[b * SDIM + i];
    red[t] = s;
    __syncthreads();
    for (int off = 128; off > 0; off >>= 1) {
        if (t < off) red[t] += red[t + off];
        __syncthreads();
    }
    if (t == 0) len[b] = red[0];
}

// --- fused mask + GEMM + bias + context emit --------------------------------
// Block: 256 threads = 8 waves. Wave w owns output columns [16w, 16w+16).
// NT = number of fully in-bounds 16-row tiles this block owns (4, or 3 for
// the single tail block). All loads/stores are unconditional; masking is
// applied only by wave 0 (for the context output) and in the epilogue select.
template <int NT>
__device__ __forceinline__ void head_body(const float* __restrict__ x,
                                          const float* __restrict__ W,
                                          const float* __restrict__ bias,
                                          const int*   __restrict__ len,
                                          float* __restrict__ logits,
                                          float* __restrict__ ctx,
                                          int mbase) {
    const int lane = threadIdx.x & 31;
    const int wave = threadIdx.x >> 5;
    const int half = lane >> 4;          // selects K pair {0,1} vs {2,3}
    const int l16  = lane & 15;
    const int n    = wave * 16 + l16;    // output column (0..127)

    // Per-lane A-row / context-row pointers (rows guaranteed in-bounds).
    const float* ap[NT];
    float*       cp[NT];
#pragma unroll
    for (int t = 0; t < NT; ++t) {
        int m  = mbase + t * 16 + l16;
        int bb = m / SM2;
        int ss = m % SM2;
        ap[t] = x + ((size_t)bb * SDIM + (ss + 1)) * HDIM + 2 * half; // x[b,s+1]
        cp[t] = ctx + (size_t)m * HDIM + 2 * half;
    }
    const float* wp = W + (size_t)n * HDIM + 2 * half;  // B[k][n] = W[n][k]

    v8f acc[NT] = {};

    if (wave == 0) {
        // Wave 0 emits the masked context, so it scales fragments to zero for
        // trimmed rows before both the store and the WMMA.
        float sc[NT];
#pragma unroll
        for (int t = 0; t < NT; ++t) {
            int m  = mbase + t * 16 + l16;
            int bb = m / SM2;
            int ss = m % SM2;
            sc[t] = (ss < len[bb] - 2) ? 1.0f : 0.0f;
        }
#pragma unroll 4
        for (int k0 = 0; k0 < HDIM; k0 += 4) {
            v2f bf = *(const v2f*)(wp + k0);
            v2f a[NT];
#pragma unroll
            for (int t = 0; t < NT; ++t) a[t] = *(const v2f*)(ap[t] + k0);
#pragma unroll
            for (int t = 0; t < NT; ++t) {
                v2f am = a[t] * sc[t];
                *(v2f*)(cp[t] + k0) = am;           // 16B/row/step, streaming
                acc[t] = __builtin_amdgcn_wmma_f32_16x16x4_f32(
                    false, am, false, bf, (short)0, acc[t], false, false);
            }
        }
    } else {
        // Hot path (7 of 8 waves): no masking in-loop. Invalid rows accumulate
        // garbage that the epilogue discards (reference gives them logits = b).
#pragma unroll 4
        for (int k0 = 0; k0 < HDIM; k0 += 4) {
            v2f bf = *(const v2f*)(wp + k0);
            v2f a[NT];
#pragma unroll
            for (int t = 0; t < NT; ++t) a[t] = *(const v2f*)(ap[t] + k0);
#pragma unroll
            for (int t = 0; t < NT; ++t) {
                acc[t] = __builtin_amdgcn_wmma_f32_16x16x4_f32(
                    false, a[t], false, bf, (short)0, acc[t], false, false);
            }
        }
    }

    // Epilogue: bias add + validity select + store.
    // acc vgpr i holds row (i + 8*half) of the 16x16 tile, column n.
    const float bv = bias[n];
#pragma unroll
    for (int t = 0; t < NT; ++t) {
#pragma unroll
        for (int i = 0; i < 8; ++i) {
            int  m  = mbase + t * 16 + half * 8 + i;
            int  bb = m / SM2;
            int  ss = m % SM2;
            bool v  = ss < len[bb] - 2;
            logits[(size_t)m * LDIM + n] = v ? (acc[t][i] + bv) : bv;
        }
    }
}

__global__ __launch_bounds__(256)
void head_kernel(const float* __restrict__ x,
                 const float* __restrict__ W,
                 const float* __restrict__ bias,
                 const int*   __restrict__ len,
                 float* __restrict__ logits,
                 float* __restrict__ ctx) {
    const int mbase = blockIdx.x * MTILE;
    if (mbase + MTILE <= MROWS) {
        head_body<4>(x, W, bias, len, logits, ctx, mbase);
    } else {
        // Tail block: MROWS - mbase == 48 == 3 whole 16-row tiles.
        head_body<3>(x, W, bias, len, logits, ctx, mbase);
    }
}

// ---------------------------------------------------------------------------
extern "C" void kernel_launch(void* const* d_in, const int* in_sizes, int n_in,
                              void* d_out, int out_size, void* d_ws, size_t ws_size,
                              hipStream_t stream) {
    const float* x    = (const float*)d_in[0];
    const int*   mask = (const int*)  d_in[1];
    const float* W    = (const float*)d_in[2];
    const float* bias = (const float*)d_in[3];

    float* logits = (float*)d_out;                          // (B, S-2, L)
    float* ctx    = logits + (size_t)BSZ * SM2 * LDIM;      // (B, S-2, H)
    int*   len    = (int*)d_ws;                             // (B,)

    seqlen_kernel<<<BSZ, 256, 0, stream>>>(mask, len);

    const int grid = (MROWS + MTILE - 1) / MTILE;           // 512 workgroups
    head_kernel<<<grid, 256, 0, stream>>>(x, W, bias, len, logits, ctx);
}